// TopKMoEGate_68049461838425
// MI455X (gfx1250) — compile-verified
//
#include <hip/hip_runtime.h>
#include <hip/hip_bf16.h>

typedef __attribute__((ext_vector_type(2))) float v2f;
typedef __attribute__((ext_vector_type(8))) float v8f;

#define T_TOKENS 16384
#define DDIM     2048
#define EXPERTS  64
#define NEG_INF  (-3.402823466e38f)

// ---------------------------------------------------------------------------
// Prep: repack W_gate [E, D] into paired-K layout for WMMA B fragments:
//   Wt2[(d>>1)*2*E + e*2 + (d&1)] = W[e*D + d]
// so a lane's (K=k, K=k+1) pair for column n is one contiguous 8-byte load.
// Also zeroes the 64-entry load-balance accumulator.
// ---------------------------------------------------------------------------
__global__ void moe_prep_kernel(const float* __restrict__ W,
                                float* __restrict__ Wt2,
                                float* __restrict__ acc) {
    int tid = blockIdx.x * blockDim.x + threadIdx.x;
    if (tid < DDIM * EXPERTS) {
        int d = tid >> 6;       // 0..2047
        int e = tid & 63;       // 0..63
        Wt2[(size_t)(d >> 1) * (2 * EXPERTS) + e * 2 + (d & 1)] =
            W[(size_t)e * DDIM + d];
    }
    if (tid < EXPERTS) acc[tid] = 0.0f;
}

// ---------------------------------------------------------------------------
// Main fused kernel: 64 threads = 2 waves per block, 16 rows per wave.
// Each wave: logits[16x64] via V_WMMA_F32_16X16X4_F32 (4 N-tiles), then
// noisy top-2 + softmax + load-balance accumulation.
// ---------------------------------------------------------------------------
__global__ void __launch_bounds__(64)
moe_gate_kernel(const float* __restrict__ x,
                const float* __restrict__ noise,
                const float* __restrict__ noise_weight,
                const float* __restrict__ Wt2,
                float* __restrict__ out_gates,
                float* __restrict__ out_ids,
                float* __restrict__ acc) {
    __shared__ float stage[2][16 * 64];   // logits, then reused for gates
    __shared__ float noiseT[2][16 * 64];
    __shared__ float lb[EXPERTS];
    __shared__ float nwS[EXPERTS];

    const int tid  = threadIdx.x;
    const int wave = tid >> 5;
    const int lane = tid & 31;
    const int hi   = lane >> 4;     // 0 or 1 (lane half)
    const int nl   = lane & 15;

    lb[tid]  = 0.0f;                // blockDim == 64 == EXPERTS
    nwS[tid] = noise_weight[tid];
    __syncthreads();

    const int t0w = blockIdx.x * 32 + wave * 16;   // first row of this wave

    // ---- GEMM: 16 rows x 64 experts, K-loop over D in steps of 4 ----
    // A (16x4 f32): lane<16 -> M=lane, K = {0,1}; lane>=16 -> M=lane-16, K={2,3}
    const float* pA = x + (size_t)(t0w + nl) * DDIM + 2 * hi;
    // B (4x16 f32): vgpr j holds K = j + 2*hi, N = nl; paired-K repack -> b64
    const float* pB = Wt2 + (size_t)hi * (2 * EXPERTS) + nl * 2;

    v8f c0 = {0.f,0.f,0.f,0.f,0.f,0.f,0.f,0.f};
    v8f c1 = c0, c2 = c0, c3 = c0;

#pragma unroll 4
    for (int k = 0; k < DDIM; k += 4) {
        v2f a = *(const v2f*)(pA + k);
        const float* pb = pB + (size_t)(k >> 1) * (2 * EXPERTS);
        v2f b0 = *(const v2f*)(pb);
        v2f b1 = *(const v2f*)(pb + 32);
        v2f b2 = *(const v2f*)(pb + 64);
        v2f b3 = *(const v2f*)(pb + 96);
        c0 = __builtin_amdgcn_wmma_f32_16x16x4_f32(false, a, false, b0, (short)0, c0, false, false);
        c1 = __builtin_amdgcn_wmma_f32_16x16x4_f32(false, a, false, b1, (short)0, c1, false, false);
        c2 = __builtin_amdgcn_wmma_f32_16x16x4_f32(false, a, false, b2, (short)0, c2, false, false);
        c3 = __builtin_amdgcn_wmma_f32_16x16x4_f32(false, a, false, b3, (short)0, c3, false, false);
    }

    // ---- C tiles -> LDS: vgpr j holds row (j + 8*hi), col nl of each tile ----
    float* st = stage[wave];
#pragma unroll
    for (int j = 0; j < 8; ++j) {
        int row = j + 8 * hi;
        st[row * 64 +  0 + nl] = c0[j];
        st[row * 64 + 16 + nl] = c1[j];
        st[row * 64 + 32 + nl] = c2[j];
        st[row * 64 + 48 + nl] = c3[j];
    }

    // ---- stage the 16x64 noise tile (coalesced float4) ----
    {
        const float4* ng = (const float4*)(noise + (size_t)t0w * EXPERTS);
        float4* nt = (float4*)noiseT[wave];
#pragma unroll
        for (int i = 0; i < 8; ++i) nt[i * 32 + lane] = ng[i * 32 + lane];
    }
    __syncthreads();

    // ---- per-row: noisy top-2, softmax(top-2), clean softmax accumulation ----
    if (lane < 16) {
        const int r = lane;
        const int t = t0w + r;
        float* lrow = st + r * 64;
        const float* nrow = noiseT[wave] + r * 64;

        float maxc = NEG_INF;
        float v1 = NEG_INF, v2 = NEG_INF;
        int   i1 = 0, i2 = 0;
#pragma unroll 4
        for (int e = 0; e < EXPERTS; ++e) {
            float l  = lrow[e];
            maxc = fmaxf(maxc, l);
            float ln = fmaf(nrow[e], nwS[e], l);   // NOISY_STD == 1.0
            if (ln > v1)      { v2 = v1; i2 = i1; v1 = ln; i1 = e; }
            else if (ln > v2) { v2 = ln; i2 = e; }
        }
        float s = 0.0f;
#pragma unroll 4
        for (int e = 0; e < EXPERTS; ++e) s += __expf(lrow[e] - maxc);
        const float inv = 1.0f / s;

        // softmax over the two selected noisy logits (v1 >= v2)
        const float e2 = __expf(v2 - v1);
        const float g1 = 1.0f / (1.0f + e2);
        const float g2 = e2 * g1;

#pragma unroll 4
        for (int e = 0; e < EXPERTS; ++e) {
            float l = lrow[e];
            atomicAdd(&lb[e], __expf(l - maxc) * inv);
            float gv = (e == i1) ? g1 : ((e == i2) ? g2 : 0.0f);
            lrow[e] = gv;   // reuse stage as gate output
        }
        out_ids[(size_t)t * 2 + 0] = (float)i1;
        out_ids[(size_t)t * 2 + 1] = (float)i2;
    }
    __syncthreads();

    // ---- coalesced gate output store (float4) ----
    {
        float4* og = (float4*)(out_gates + (size_t)t0w * EXPERTS);
        const float4* sg = (const float4*)st;
#pragma unroll
        for (int i = 0; i < 8; ++i) og[i * 32 + lane] = sg[i * 32 + lane];
    }

    // ---- per-block load-balance partial -> global accumulator ----
    __syncthreads();
    atomicAdd(&acc[tid], lb[tid]);
}

// ---------------------------------------------------------------------------
// Finalize: loss = mean((acc/T - 1/E)^2) * 0.01
// ---------------------------------------------------------------------------
__global__ void moe_finalize_kernel(const float* __restrict__ acc,
                                    float* __restrict__ out_loss) {
    __shared__ float sh[EXPERTS];
    const int t = threadIdx.x;
    float m = acc[t] * (1.0f / (float)T_TOKENS) - (1.0f / (float)EXPERTS);
    sh[t] = m * m;
    __syncthreads();
    for (int s = EXPERTS / 2; s > 0; s >>= 1) {
        if (t < s) sh[t] += sh[t + s];
        __syncthreads();
    }
    if (t == 0) *out_loss = sh[0] * (1.0f / (float)EXPERTS) * 0.01f;
}

extern "C" void kernel_launch(void* const* d_in, const int* in_sizes, int n_in,
                              void* d_out, int out_size, void* d_ws, size_t ws_size,
                              hipStream_t stream) {
    const float* x     = (const float*)d_in[0];  // [T, D]
    const float* noise = (const float*)d_in[1];  // [T, E]
    const float* W     = (const float*)d_in[2];  // [E, D]
    const float* nw    = (const float*)d_in[3];  // [E]
    // d_in[4] is k == 2 (hardcoded)

    float* ws   = (float*)d_ws;
    float* Wt2  = ws;                      // D*E floats (512 KB)
    float* acc  = ws + (size_t)DDIM * EXPERTS;  // E floats

    float* out_gates = (float*)d_out;                               // T*E
    float* out_ids   = out_gates + (size_t)T_TOKENS * EXPERTS;      // T*2
    float* out_loss  = out_ids + (size_t)T_TOKENS * 2;              // 1

    moe_prep_kernel<<<(DDIM * EXPERTS + 255) / 256, 256, 0, stream>>>(W, Wt2, acc);
    moe_gate_kernel<<<T_TOKENS / 32, 64, 0, stream>>>(x, noise, nw, Wt2,
                                                      out_gates, out_ids, acc);
    moe_finalize_kernel<<<1, EXPERTS, 0, stream>>>(acc, out_loss);
}